// GATModel_7370163880476
// MI455X (gfx1250) — compile-verified
//
#include <hip/hip_runtime.h>
#include <math.h>

typedef __attribute__((ext_vector_type(2))) float v2f;
typedef __attribute__((ext_vector_type(8))) float v8f;

#define HEADS 4
#define NEG_SLOPE 0.2f

// ---------------------------------------------------------------------------
// WMMA fp32 GEMM: C[M,NOUT] = A[M,K] x B[K,NOUT].  One wave per 16x16 tile.
// Uses V_WMMA_F32_16X16X4_F32 (exact fp32 accumulate).
// A 16x4 fragment layout (ISA 7.12.2): lanes 0-15 hold M=lane, K=(0,1) in
// (v0,v1); lanes 16-31 hold M=lane-16, K=(2,3). B 4x16 mirrors it. C/D:
// VGPR r, lanes 0-15 -> row r; lanes 16-31 -> row r+8.
// ---------------------------------------------------------------------------
template<int K, int NOUT>
__global__ __launch_bounds__(256)
void gemm_wmma_f32(const float* __restrict__ A, const float* __restrict__ B,
                   float* __restrict__ C, int M)
{
    const int wavesPerBlock = 8;                     // 256 threads / wave32
    int wave = blockIdx.x * wavesPerBlock + (threadIdx.x >> 5);
    int lane = threadIdx.x & 31;
    const int ntiles = NOUT / 16;
    const int mtiles = M / 16;
    if (wave >= mtiles * ntiles) return;
    int m0 = (wave / ntiles) * 16;
    int n0 = (wave % ntiles) * 16;

    int l  = lane & 15;
    int kh = lane >> 4;                              // 0 => K pair (0,1), 1 => (2,3)
    const float* arow = A + (size_t)(m0 + l) * K + 2 * kh;

    v8f acc = {};
    #pragma unroll 4
    for (int k = 0; k < K; k += 4) {
        v2f a, b;
        a.x = arow[k + 0];
        a.y = arow[k + 1];
        b.x = B[(size_t)(k + 2 * kh + 0) * NOUT + n0 + l];
        b.y = B[(size_t)(k + 2 * kh + 1) * NOUT + n0 + l];
        acc = __builtin_amdgcn_wmma_f32_16x16x4_f32(
            /*neg_a=*/false, a, /*neg_b=*/false, b,
            /*c_mod=*/(short)0, acc, /*reuse_a=*/false, /*reuse_b=*/false);
    }
    #pragma unroll
    for (int r = 0; r < 8; ++r) {
        C[(size_t)(m0 + 8 * kh + r) * NOUT + n0 + l] = acc[r];
    }
}

// ---------------------------------------------------------------------------
// alpha_s[n,h] = <h[n,h,:], a_s[h,:]>, same for a_d.  One thread per (n,h).
// ---------------------------------------------------------------------------
template<int C>
__global__ void alpha_kernel(const float* __restrict__ h,
                             const float* __restrict__ a_s,
                             const float* __restrict__ a_d,
                             float* __restrict__ as_out,
                             float* __restrict__ ad_out, int N)
{
    int idx = blockIdx.x * blockDim.x + threadIdx.x;     // n*HEADS + hh
    if (idx >= N * HEADS) return;
    int n = idx / HEADS, hh = idx % HEADS;
    const float* hp  = h + (size_t)n * (HEADS * C) + hh * C;
    const float* asp = a_s + hh * C;
    const float* adp = a_d + hh * C;
    float s = 0.f, d = 0.f;
    #pragma unroll 8
    for (int c = 0; c < C; ++c) { s += hp[c] * asp[c]; d += hp[c] * adp[c]; }
    as_out[idx] = s;
    ad_out[idx] = d;
}

__global__ void fill_f32(float* __restrict__ p, float v, int n)
{
    int i = blockIdx.x * blockDim.x + threadIdx.x;
    if (i < n) p[i] = v;
}

__device__ __forceinline__ void edge_sd(const int* __restrict__ ei, int E,
                                        int e, int& s, int& d)
{
    if (e < E) { s = ei[e]; d = ei[E + e]; }
    else       { s = e - E; d = e - E; }                 // self loops appended
}

__device__ __forceinline__ float leaky(float v)
{
    return v > 0.f ? v : v * NEG_SLOPE;
}

// float atomic max via int/uint ordering trick (init value must be -inf bits)
__device__ __forceinline__ void atomicMaxF(float* addr, float v)
{
    if (v >= 0.f) {
        if (v == 0.f) v = 0.f;                           // canonicalize -0
        atomicMax((int*)addr, __float_as_int(v));
    } else {
        atomicMin((unsigned int*)addr, __float_as_uint(v));
    }
}

// pass 1: segment max over incoming edges, per (edge, head)
__global__ void edge_max_kernel(const int* __restrict__ ei, int E, int N,
                                const float* __restrict__ as,
                                const float* __restrict__ ad,
                                float* __restrict__ m)
{
    int t = blockIdx.x * blockDim.x + threadIdx.x;
    int Etot = E + N;
    if (t >= Etot * HEADS) return;
    int e = t / HEADS, hh = t % HEADS;
    int s, d; edge_sd(ei, E, e, s, d);
    float v = leaky(as[s * HEADS + hh] + ad[d * HEADS + hh]);
    atomicMaxF(&m[d * HEADS + hh], v);
}

// pass 2: denom[dst,h] += exp(e - m[dst,h])
__global__ void edge_denom_kernel(const int* __restrict__ ei, int E, int N,
                                  const float* __restrict__ as,
                                  const float* __restrict__ ad,
                                  const float* __restrict__ m,
                                  float* __restrict__ denom)
{
    int t = blockIdx.x * blockDim.x + threadIdx.x;
    int Etot = E + N;
    if (t >= Etot * HEADS) return;
    int e = t / HEADS, hh = t % HEADS;
    int s, d; edge_sd(ei, E, e, s, d);
    float v = leaky(as[s * HEADS + hh] + ad[d * HEADS + hh]);
    atomicAdd(&denom[d * HEADS + hh], expf(v - m[d * HEADS + hh]));
}

// pass 3: out[dst,h,c] += softmax_weight(e,h) * h[src,h,c].  Wave per edge.
template<int C>
__global__ __launch_bounds__(256)
void edge_aggr_kernel(const int* __restrict__ ei, int E, int N,
                      const float* __restrict__ h,
                      const float* __restrict__ as,
                      const float* __restrict__ ad,
                      const float* __restrict__ m,
                      const float* __restrict__ denom,
                      float* __restrict__ out)
{
    const int F = HEADS * C;
    int wave = blockIdx.x * (blockDim.x >> 5) + (threadIdx.x >> 5);
    int lane = threadIdx.x & 31;
    int Etot = E + N;
    if (wave >= Etot) return;
    int s, d; edge_sd(ei, E, wave, s, d);

    float w[HEADS];
    #pragma unroll
    for (int hh = 0; hh < HEADS; ++hh) {
        float v  = leaky(as[s * HEADS + hh] + ad[d * HEADS + hh]);
        float ex = expf(v - m[d * HEADS + hh]);
        w[hh] = ex / (denom[d * HEADS + hh] + 1e-16f);
    }
    const float* hs = h + (size_t)s * F;
    float*       od = out + (size_t)d * F;
    #pragma unroll
    for (int f = lane; f < F; f += 32) {
        atomicAdd(&od[f], w[f / C] * hs[f]);
    }
}

__global__ void bias_elu_kernel(float* __restrict__ out,
                                const float* __restrict__ b, int N, int F)
{
    int t = blockIdx.x * blockDim.x + threadIdx.x;
    if (t >= N * F) return;
    float v = out[t] + b[t % F];
    out[t] = v > 0.f ? v : (expf(v) - 1.f);
}

// layer 3: mean over heads (4 x 32) + bias -> z[N,32]
__global__ void mean_bias_kernel(const float* __restrict__ agg,
                                 const float* __restrict__ b3,
                                 float* __restrict__ z, int N)
{
    int t = blockIdx.x * blockDim.x + threadIdx.x;
    if (t >= N * 32) return;
    int n = t >> 5, c = t & 31;
    const float* p = agg + (size_t)n * 128;
    z[t] = 0.25f * (p[c] + p[32 + c] + p[64 + c] + p[96 + c]) + b3[c];
}

// logits[e] = <z[s], z[d]>.  Wave per candidate edge, lane = feature.
__global__ __launch_bounds__(256)
void logits_kernel(const int* __restrict__ eli, int EC,
                   const float* __restrict__ z, float* __restrict__ out)
{
    int wave = blockIdx.x * (blockDim.x >> 5) + (threadIdx.x >> 5);
    int lane = threadIdx.x & 31;
    if (wave >= EC) return;
    int s = eli[wave], d = eli[EC + wave];
    float p = z[(size_t)s * 32 + lane] * z[(size_t)d * 32 + lane];
    #pragma unroll
    for (int off = 16; off; off >>= 1) p += __shfl_down(p, off, 32);
    if (lane == 0) out[wave] = p;
}

// ---------------------------------------------------------------------------
static inline int cdiv(long long a, long long b) { return (int)((a + b - 1) / b); }

extern "C" void kernel_launch(void* const* d_in, const int* in_sizes, int n_in,
                              void* d_out, int out_size, void* d_ws, size_t ws_size,
                              hipStream_t stream)
{
    const float* x   = (const float*)d_in[0];
    const int*   ei  = (const int*)  d_in[1];
    const int*   eli = (const int*)  d_in[2];
    const float* W1  = (const float*)d_in[3];
    const float* a1s = (const float*)d_in[4];
    const float* a1d = (const float*)d_in[5];
    const float* b1  = (const float*)d_in[6];
    const float* W2  = (const float*)d_in[7];
    const float* a2s = (const float*)d_in[8];
    const float* a2d = (const float*)d_in[9];
    const float* b2  = (const float*)d_in[10];
    const float* W3  = (const float*)d_in[11];
    const float* a3s = (const float*)d_in[12];
    const float* a3d = (const float*)d_in[13];
    const float* b3  = (const float*)d_in[14];

    const int N  = in_sizes[0] / 384;      // 50000
    const int E  = in_sizes[1] / 2;        // 800000
    const int EC = in_sizes[2] / 2;        // 200000
    const int Etot = E + N;

    // workspace layout (floats)
    float* ws  = (float*)d_ws;
    float* h   = ws;                        // [N, 256] GEMM output (max width)
    float* agg = h   + (size_t)N * 256;     // [N, 256] aggregated output
    float* as_ = agg + (size_t)N * 256;     // [N, 4]
    float* ad_ = as_ + (size_t)N * HEADS;   // [N, 4]
    float* mx  = ad_ + (size_t)N * HEADS;   // [N, 4]
    float* dn  = mx  + (size_t)N * HEADS;   // [N, 4]
    float* z   = dn  + (size_t)N * HEADS;   // [N, 32]

    const int TB = 256;
    const int nh = N * HEADS;

    // ---------------- layer 1: 384 -> 4x64 (concat 256) ----------------
    gemm_wmma_f32<384, 256><<<cdiv((size_t)(N / 16) * 16, 8), TB, 0, stream>>>(x, W1, h, N);
    alpha_kernel<64><<<cdiv(nh, TB), TB, 0, stream>>>(h, a1s, a1d, as_, ad_, N);
    fill_f32<<<cdiv(nh, TB), TB, 0, stream>>>(mx, -INFINITY, nh);
    hipMemsetAsync(dn, 0, (size_t)nh * 4, stream);
    hipMemsetAsync(agg, 0, (size_t)N * 256 * 4, stream);
    edge_max_kernel  <<<cdiv((long long)Etot * HEADS, TB), TB, 0, stream>>>(ei, E, N, as_, ad_, mx);
    edge_denom_kernel<<<cdiv((long long)Etot * HEADS, TB), TB, 0, stream>>>(ei, E, N, as_, ad_, mx, dn);
    edge_aggr_kernel<64><<<cdiv(Etot, 8), TB, 0, stream>>>(ei, E, N, h, as_, ad_, mx, dn, agg);
    bias_elu_kernel<<<cdiv((long long)N * 256, TB), TB, 0, stream>>>(agg, b1, N, 256);

    // ---------------- layer 2: 256 -> 4x16 (concat 64) ----------------
    gemm_wmma_f32<256, 64><<<cdiv((size_t)(N / 16) * 4, 8), TB, 0, stream>>>(agg, W2, h, N);
    alpha_kernel<16><<<cdiv(nh, TB), TB, 0, stream>>>(h, a2s, a2d, as_, ad_, N);
    fill_f32<<<cdiv(nh, TB), TB, 0, stream>>>(mx, -INFINITY, nh);
    hipMemsetAsync(dn, 0, (size_t)nh * 4, stream);
    hipMemsetAsync(agg, 0, (size_t)N * 64 * 4, stream);
    edge_max_kernel  <<<cdiv((long long)Etot * HEADS, TB), TB, 0, stream>>>(ei, E, N, as_, ad_, mx);
    edge_denom_kernel<<<cdiv((long long)Etot * HEADS, TB), TB, 0, stream>>>(ei, E, N, as_, ad_, mx, dn);
    edge_aggr_kernel<16><<<cdiv(Etot, 8), TB, 0, stream>>>(ei, E, N, h, as_, ad_, mx, dn, agg);
    bias_elu_kernel<<<cdiv((long long)N * 64, TB), TB, 0, stream>>>(agg, b2, N, 64);

    // ---------------- layer 3: 64 -> 4x32 (mean 32) ----------------
    gemm_wmma_f32<64, 128><<<cdiv((size_t)(N / 16) * 8, 8), TB, 0, stream>>>(agg, W3, h, N);
    alpha_kernel<32><<<cdiv(nh, TB), TB, 0, stream>>>(h, a3s, a3d, as_, ad_, N);
    fill_f32<<<cdiv(nh, TB), TB, 0, stream>>>(mx, -INFINITY, nh);
    hipMemsetAsync(dn, 0, (size_t)nh * 4, stream);
    hipMemsetAsync(agg, 0, (size_t)N * 128 * 4, stream);
    edge_max_kernel  <<<cdiv((long long)Etot * HEADS, TB), TB, 0, stream>>>(ei, E, N, as_, ad_, mx);
    edge_denom_kernel<<<cdiv((long long)Etot * HEADS, TB), TB, 0, stream>>>(ei, E, N, as_, ad_, mx, dn);
    edge_aggr_kernel<32><<<cdiv(Etot, 8), TB, 0, stream>>>(ei, E, N, h, as_, ad_, mx, dn, agg);
    mean_bias_kernel<<<cdiv((long long)N * 32, TB), TB, 0, stream>>>(agg, b3, z, N);

    // ---------------- readout ----------------
    logits_kernel<<<cdiv(EC, 8), TB, 0, stream>>>(eli, EC, z, (float*)d_out);
}